// UnsupervisedDiffusionLoss_30623116820897
// MI455X (gfx1250) — compile-verified
//
#include <hip/hip_runtime.h>
#include <hip/hip_bf16.h>
#include <math.h>

#define BB 8
#define NN 4096
#define DCONT 1024
#define DSTYLE 256

typedef float v2f __attribute__((ext_vector_type(2)));
typedef float v8f __attribute__((ext_vector_type(8)));

// ---------------------------------------------------------------------------
// Wave32 sum-reduction through the matrix pipe (exact f32 math).
// A (16x4 f32): VGPR0 lanes0-15 -> A[m][0]=p_m, lanes16-31 -> A[m][2]=p_{m+16};
// VGPR1 = 0.  B = ones(4x16).  D[m][n] = p_m + p_{m+16} for every n.
// Lane n<16 holds rows 0..7 of D in its 8 VGPRs; lane n>=16 holds rows 8..15.
// 8 VALU adds + one xor-16 shuffle complete the 32-lane sum.
// EXEC must be all-ones at every call site (all calls are block-uniform).
// ---------------------------------------------------------------------------
__device__ __forceinline__ float wave_sum_wmma(float v) {
  v2f a; a[0] = v;    a[1] = 0.0f;
  v2f b; b[0] = 1.0f; b[1] = 1.0f;
  v8f c = {0.f, 0.f, 0.f, 0.f, 0.f, 0.f, 0.f, 0.f};
  c = __builtin_amdgcn_wmma_f32_16x16x4_f32(false, a, false, b, (short)0, c,
                                            false, false);
  float s = c[0] + c[1] + c[2] + c[3] + c[4] + c[5] + c[6] + c[7];
  s += __shfl_xor(s, 16, 32);
  return s;
}

// ---------------------------------------------------------------------------
// Workspace layout (floats):
//  [0]      sum (pred-target)^2
//  [1]      sum (content_noisy-content_orig)^2
//  [2..5]   gz_sum, gz_sumsq, oz_sum, oz_sumsq
//  [6..13]  per-batch chamfer dir0 sum-of-mins  (query=gen, db=orig)
//  [14..21] per-batch chamfer dir1 sum-of-mins  (query=orig, db=gen)
//  [22..29] per-batch seg_len sum
//  [30..37] per-batch seg_len sumsq
// ---------------------------------------------------------------------------
__global__ void zero_ws_kernel(float* ws) {
  if (threadIdx.x < 64) ws[threadIdx.x] = 0.0f;
}

__global__ void sq_diff_sum_kernel(const float* __restrict__ a,
                                   const float* __restrict__ b, int n,
                                   float* __restrict__ out) {
  float acc = 0.0f;
  for (int i = blockIdx.x * blockDim.x + threadIdx.x; i < n;
       i += gridDim.x * blockDim.x) {
    float d = a[i] - b[i];
    acc += d * d;
  }
  float s = wave_sum_wmma(acc);
  if ((threadIdx.x & 31) == 0) atomicAdd(out, s);
}

__global__ void zstats_kernel(const float* __restrict__ g,
                              const float* __restrict__ o,
                              float* __restrict__ ws) {
  float gs = 0.f, gss = 0.f, os = 0.f, oss = 0.f;
  const int n = BB * NN;
  for (int i = blockIdx.x * blockDim.x + threadIdx.x; i < n;
       i += gridDim.x * blockDim.x) {
    float z = g[i * 3 + 2];
    float w = o[i * 3 + 2];
    gs += z; gss += z * z;
    os += w; oss += w * w;
  }
  float r0 = wave_sum_wmma(gs);
  float r1 = wave_sum_wmma(gss);
  float r2 = wave_sum_wmma(os);
  float r3 = wave_sum_wmma(oss);
  if ((threadIdx.x & 31) == 0) {
    atomicAdd(&ws[2], r0);
    atomicAdd(&ws[3], r1);
    atomicAdd(&ws[4], r2);
    atomicAdd(&ws[5], r3);
  }
}

// grid = (NN/256, BB, 2); block = 256.  LDS holds the full 4096x3 db cloud.
// Inner loop: 4 points per iteration via 3x ds_load_b128 (broadcast reads,
// conflict-free), 4 independent min accumulators for dual-issue VALU ILP.
__global__ void chamfer_kernel(const float* __restrict__ gen,
                               const float* __restrict__ org,
                               float* __restrict__ ws) {
  const int b   = blockIdx.y;
  const int dir = blockIdx.z;
  const float* q  = (dir == 0 ? gen : org) + (size_t)b * NN * 3;
  const float* db = (dir == 0 ? org : gen) + (size_t)b * NN * 3;

  __shared__ float sdb[NN * 3];  // 48 KB of the 320 KB/WGP
  {
    const float4* src = (const float4*)db;
    float4* dst = (float4*)sdb;
    for (int i = threadIdx.x; i < (NN * 3) / 4; i += blockDim.x)
      dst[i] = src[i];
  }
  __syncthreads();

  const int qi = blockIdx.x * blockDim.x + threadIdx.x;
  const float qx = q[qi * 3 + 0];
  const float qy = q[qi * 3 + 1];
  const float qz = q[qi * 3 + 2];

  float m0 = 3.402823466e+38f, m1 = m0, m2 = m0, m3 = m0;
  const float4* s4 = (const float4*)sdb;
#pragma unroll 2
  for (int j = 0; j < NN / 4; ++j) {
    float4 a0 = s4[j * 3 + 0];  // x0 y0 z0 x1
    float4 a1 = s4[j * 3 + 1];  // y1 z1 x2 y2
    float4 a2 = s4[j * 3 + 2];  // z2 x3 y3 z3
    float d0 = fabsf(qx - a0.x) + fabsf(qy - a0.y) + fabsf(qz - a0.z);
    float d1 = fabsf(qx - a0.w) + fabsf(qy - a1.x) + fabsf(qz - a1.y);
    float d2 = fabsf(qx - a1.z) + fabsf(qy - a1.w) + fabsf(qz - a2.x);
    float d3 = fabsf(qx - a2.y) + fabsf(qy - a2.z) + fabsf(qz - a2.w);
    m0 = fminf(m0, d0);
    m1 = fminf(m1, d1);
    m2 = fminf(m2, d2);
    m3 = fminf(m3, d3);
  }
  float mind = fminf(fminf(m0, m1), fminf(m2, m3));

  float s = wave_sum_wmma(mind);
  if ((threadIdx.x & 31) == 0)
    atomicAdd(&ws[(dir ? 14 : 6) + b], s);
}

// grid = BB, block = 256: per-batch segment-length sum & sumsq.
__global__ void smooth_kernel(const float* __restrict__ g,
                              float* __restrict__ ws) {
  const int b = blockIdx.x;
  const float* p = g + (size_t)b * NN * 3;
  float s = 0.f, ss = 0.f;
  for (int j = threadIdx.x; j < NN - 1; j += blockDim.x) {
    float dx = p[(j + 1) * 3 + 0] - p[j * 3 + 0];
    float dy = p[(j + 1) * 3 + 1] - p[j * 3 + 1];
    float dz = p[(j + 1) * 3 + 2] - p[j * 3 + 2];
    float len = sqrtf(dx * dx + dy * dy + dz * dz);
    s += len;
    ss += len * len;
  }
  float rs  = wave_sum_wmma(s);
  float rss = wave_sum_wmma(ss);
  if ((threadIdx.x & 31) == 0) {
    atomicAdd(&ws[22 + b], rs);
    atomicAdd(&ws[30 + b], rss);
  }
}

__global__ void finalize_kernel(const float* __restrict__ ws,
                                const int* __restrict__ epoch_p,
                                float* __restrict__ out) {
  if (threadIdx.x != 0 || blockIdx.x != 0) return;

  const float l_diff    = ws[0] / (float)(BB * NN * 3);
  const float l_content = ws[1] / (float)(BB * DCONT);

  const float nz = (float)(BB * NN);
  float gm = ws[2] / nz, om = ws[4] / nz;
  float gv = (ws[3] - nz * gm * gm) / (nz - 1.0f);
  float ov = (ws[5] - nz * om * om) / (nz - 1.0f);
  const float l_lidar =
      fabsf(gm - om) +
      fabsf(sqrtf(fmaxf(gv, 0.f)) - sqrtf(fmaxf(ov, 0.f)));

  float l_ch = 0.f;
  for (int b = 0; b < BB; ++b) {
    float d1 = ws[6 + b]  / (float)NN;
    float d2 = ws[14 + b] / (float)NN;
    l_ch += 0.5f * (d1 + d2);
  }
  l_ch /= (float)BB;

  float l_sm = 0.f;
  const float ns = (float)(NN - 1);
  for (int b = 0; b < BB; ++b) {
    float m = ws[22 + b] / ns;
    float v = (ws[30 + b] - ns * m * m) / (ns - 1.0f);
    l_sm += sqrtf(fmaxf(v, 0.f));
  }
  l_sm /= (float)BB;

  const int e = epoch_p[0];
  float eff = 0.0f;
  if (e >= 10) {
    float prog = fminf(1.0f, (float)(e - 10) / 20.0f);
    eff = 10.0f * prog;
  }

  out[0] = l_diff + eff * l_ch + (l_content + l_lidar + 0.5f * l_sm);
}

extern "C" void kernel_launch(void* const* d_in, const int* in_sizes, int n_in,
                              void* d_out, int out_size, void* d_ws,
                              size_t ws_size, hipStream_t stream) {
  (void)in_sizes; (void)n_in; (void)out_size; (void)ws_size;
  const float* pred  = (const float*)d_in[0];
  const float* targ  = (const float*)d_in[1];
  const float* gen   = (const float*)d_in[2];
  const float* org   = (const float*)d_in[3];
  const float* corig = (const float*)d_in[4];
  const float* cnoisy= (const float*)d_in[5];
  const int*   epoch = (const int*)d_in[8];
  float* ws = (float*)d_ws;
  float* out = (float*)d_out;

  zero_ws_kernel<<<1, 64, 0, stream>>>(ws);
  sq_diff_sum_kernel<<<96, 256, 0, stream>>>(pred, targ, BB * NN * 3, ws + 0);
  sq_diff_sum_kernel<<<16, 256, 0, stream>>>(cnoisy, corig, BB * DCONT, ws + 1);
  zstats_kernel<<<32, 256, 0, stream>>>(gen, org, ws);
  dim3 gch(NN / 256, BB, 2);
  chamfer_kernel<<<gch, 256, 0, stream>>>(gen, org, ws);
  smooth_kernel<<<BB, 256, 0, stream>>>(gen, ws);
  finalize_kernel<<<1, 32, 0, stream>>>(ws, epoch, out);
}